// DistEstNet_11776800325802
// MI455X (gfx1250) — compile-verified
//
#include <hip/hip_runtime.h>
#include <hip/hip_bf16.h>

// ---------------------------------------------------------------------------
// DAGMM energy pipeline for MI455X (gfx1250), wave32 + f32 WMMA 16x16x4.
// N=65536, D=128, H=512, K=16.  All fp32 (problem is L2-resident / far from
// matrix-pipe peak, so full precision is free).
// ---------------------------------------------------------------------------

#define NN 65536
#define DD 128
#define HH 512
#define KK 16
#define EPSF 1e-12f
#define LN2PI 1.8378770664093453f

typedef float v2f __attribute__((ext_vector_type(2)));
typedef float v8f __attribute__((ext_vector_type(8)));

#if defined(__HIP_DEVICE_COMPILE__) && defined(__AMDGCN__)
#if !__has_builtin(__builtin_amdgcn_wmma_f32_16x16x4_f32)
#error "builtin __builtin_amdgcn_wmma_f32_16x16x4_f32 not available on this toolchain"
#endif
#endif

__device__ inline v8f wmma4(v2f a, v2f b, v8f c) {
#if defined(__HIP_DEVICE_COMPILE__)
  return __builtin_amdgcn_wmma_f32_16x16x4_f32(false, a, false, b, (short)0, c,
                                               false, false);
#else
  (void)a; (void)b; return c;
#endif
}

__device__ inline v8f vzero8() {
  v8f v;
#pragma unroll
  for (int i = 0; i < 8; ++i) v[i] = 0.0f;
  return v;
}

__device__ inline float block_reduce_sum(float v) {
  __shared__ float sred[8];
#pragma unroll
  for (int m = 16; m >= 1; m >>= 1) v += __shfl_xor(v, m, 32);
  int wid = threadIdx.x >> 5;
  int nw = (blockDim.x + 31) >> 5;
  if ((threadIdx.x & 31) == 0) sred[wid] = v;
  __syncthreads();
  float r = 0.0f;
  if (threadIdx.x == 0)
    for (int i = 0; i < nw; ++i) r += sred[i];
  __syncthreads();
  return r;  // valid on thread 0
}

// ---------------------------------------------------------------------------
// K0: zero the atomic accumulation targets (M2, sigma_diag)
// ---------------------------------------------------------------------------
__global__ void zero_kernel(float* M2, float* sig_diag) {
  int i = blockIdx.x * blockDim.x + threadIdx.x;
  if (i < KK * DD * DD) M2[i] = 0.0f;
  if (i == 0) *sig_diag = 0.0f;
}

// ---------------------------------------------------------------------------
// K1: gamma = sigmoid(tanh(X@W1+b1)@W2+b2).  One wave per 16-row tile.
// Layer 1 via f32 WMMA (K-loop over D=128 in steps of 4), tanh in C-layout,
// LDS transpose to A-layout, layer-2 WMMA accumulation into the gamma tile.
// ---------------------------------------------------------------------------
__global__ void gamma_kernel(const float* __restrict__ X,
                             const float* __restrict__ W1,
                             const float* __restrict__ b1,
                             const float* __restrict__ W2,
                             const float* __restrict__ b2,
                             float* __restrict__ gamma) {
  __shared__ float Xs[4][16][132];
  __shared__ float Hs[4][16][20];
  int wid = threadIdx.x >> 5;
  int lane = threadIdx.x & 31;
  int tile = blockIdx.x * 4 + wid;
  int n0 = tile * 16;
  int m = lane & 15;
  int g2 = (lane >> 4) << 1;  // 0 or 2

  // stage X tile (16x128) into LDS, padded stride 132
  for (int idx = lane; idx < 16 * DD; idx += 32)
    Xs[wid][idx >> 7][idx & 127] = X[(n0 + (idx >> 7)) * DD + (idx & 127)];

  v8f gacc = vzero8();
  for (int h0 = 0; h0 < HH; h0 += 16) {
    v8f hacc = vzero8();
#pragma unroll 4
    for (int kk = 0; kk < 32; ++kk) {
      int kb = kk * 4 + g2;
      v2f a, b;
      a.x = Xs[wid][m][kb];
      a.y = Xs[wid][m][kb + 1];
      b.x = W1[kb * HH + h0 + m];
      b.y = W1[(kb + 1) * HH + h0 + m];
      hacc = wmma4(a, b, hacc);
    }
    float bb = b1[h0 + m];
#pragma unroll
    for (int r = 0; r < 8; ++r) {
      int row = r + ((lane >> 4) << 3);
      Hs[wid][row][m] = tanhf(hacc[r] + bb);
    }
    // same-wave LDS RAW: hardware/compiler orders via DScnt
#pragma unroll
    for (int kk2 = 0; kk2 < 4; ++kk2) {
      int kb = kk2 * 4 + g2;
      v2f a, b;
      a.x = Hs[wid][m][kb];
      a.y = Hs[wid][m][kb + 1];
      b.x = W2[(h0 + kb) * KK + m];
      b.y = W2[(h0 + kb + 1) * KK + m];
      gacc = wmma4(a, b, gacc);
    }
  }
  float b2v = b2[m];
#pragma unroll
  for (int r = 0; r < 8; ++r) {
    int row = r + ((lane >> 4) << 3);
    float g = 1.0f / (1.0f + expf(-(gacc[r] + b2v)));
    gamma[(n0 + row) * KK + m] = g;
  }
}

// ---------------------------------------------------------------------------
// K2a: gamma_sum[k] ; K2b: mu_num[k][d] = sum_n gamma[n,k]*X[n,d]
// (BW-trivial reductions; plain VALU is the right tool)
// ---------------------------------------------------------------------------
__global__ void gsum_kernel(const float* __restrict__ gamma, float* gsum) {
  int k = blockIdx.x;
  float a = 0.0f;
  for (int n = threadIdx.x; n < NN; n += blockDim.x) a += gamma[n * KK + k];
  a = block_reduce_sum(a);
  if (threadIdx.x == 0) gsum[k] = a;
}

__global__ void mu_kernel(const float* __restrict__ gamma,
                          const float* __restrict__ X, float* mu_num) {
  int k = blockIdx.x >> 7;
  int d = blockIdx.x & 127;
  float a = 0.0f;
  for (int n = threadIdx.x; n < NN; n += blockDim.x)
    a += gamma[n * KK + k] * X[n * DD + d];
  a = block_reduce_sum(a);
  if (threadIdx.x == 0) mu_num[k * DD + d] = a;
}

// ---------------------------------------------------------------------------
// K3: second moment M2[k][d][e] = sum_n gamma[n,k] X[n,d] X[n,e] via WMMA.
// grid (8 d-tiles, 8 e-tiles, K*NSPLIT), one wave per block, atomic combine.
// ---------------------------------------------------------------------------
#define NSPLIT 16
__global__ void m2_kernel(const float* __restrict__ gamma,
                          const float* __restrict__ X, float* M2) {
  int d0 = blockIdx.x * 16, e0 = blockIdx.y * 16;
  int k = blockIdx.z / NSPLIT, sp = blockIdx.z % NSPLIT;
  int lane = threadIdx.x;
  int m = lane & 15;
  int g2 = (lane >> 4) << 1;
  int n0 = sp * (NN / NSPLIT);
  v8f acc = vzero8();
  for (int j = 0; j < NN / NSPLIT; j += 4) {
    int na = n0 + j + g2;
    float ga = gamma[na * KK + k];
    float gb = gamma[(na + 1) * KK + k];
    v2f a, b;
    a.x = ga * X[na * DD + d0 + m];
    a.y = gb * X[(na + 1) * DD + d0 + m];
    b.x = X[na * DD + e0 + m];
    b.y = X[(na + 1) * DD + e0 + m];
    acc = wmma4(a, b, acc);
  }
  int rbase = (lane >> 4) << 3;
#pragma unroll
  for (int r = 0; r < 8; ++r)
    atomicAdd(&M2[k * DD * DD + (d0 + rbase + r) * DD + e0 + m], acc[r]);
}

// ---------------------------------------------------------------------------
// K4a: per-k packed-lower Cholesky in LDS (33KB), log-det, sigma_diag,
// and columns of L^-1 (each thread owns one global column -> no cross-thread
// global sharing inside the block).
// ---------------------------------------------------------------------------
__global__ void chol_kernel(const float* __restrict__ M2,
                            const float* __restrict__ mu_num,
                            const float* __restrict__ gsum,
                            float* __restrict__ Linv, float* ldet_g,
                            float* sig_diag) {
  int k = blockIdx.x;
  int r = threadIdx.x;  // 128 threads
  __shared__ float Sp[DD * (DD + 1) / 2];
  __shared__ float muS[DD];
  __shared__ float ldetS;
  float gs = gsum[k];
  muS[r] = mu_num[k * DD + r] / gs;
  if (r == 0) ldetS = 0.0f;
  __syncthreads();

  const float* M2k = M2 + k * DD * DD;
  int base = r * (r + 1) / 2;
  float mur = muS[r];
  for (int c = 0; c <= r; ++c) {
    float v = M2k[r * DD + c] / gs - mur * muS[c];
    if (c == r) v += EPSF;
    Sp[base + c] = v;
  }
  float dinv = 1.0f / Sp[base + r];
  __syncthreads();
  float dsum = block_reduce_sum(dinv);
  if (r == 0) atomicAdd(sig_diag, dsum);

  for (int p = 0; p < DD; ++p) {
    int bp = p * (p + 1) / 2;
    __syncthreads();
    if (r == p) {
      float sp = sqrtf(Sp[bp + p]);
      Sp[bp + p] = sp;
      ldetS += 2.0f * logf(sp);
    }
    __syncthreads();
    if (r > p) Sp[base + p] /= Sp[bp + p];
    __syncthreads();
    if (r > p) {
      float lrp = Sp[base + p];
      for (int c = p + 1; c <= r; ++c) Sp[base + c] -= lrp * Sp[c * (c + 1) / 2 + p];
    }
  }
  __syncthreads();

  // forward solve L y = e_r  -> column r of L^-1 (thread-private global col)
  float* col = Linv + k * DD * DD + r * DD;
  col[r] = 1.0f / Sp[base + r];
  for (int t = r + 1; t < DD; ++t) {
    int bt = t * (t + 1) / 2;
    float a = 0.0f;
    for (int mm = r; mm < t; ++mm) a += Sp[bt + mm] * col[mm];
    col[t] = -a / Sp[bt + t];
  }
  if (r == 0) ldet_g[k] = ldetS;
}

// ---------------------------------------------------------------------------
// K4b: cov_inv = L^-T L^-1, scattered into Bbig[(d*128+e)*16+k]; also
// c_k[d] = (cov_inv mu)_d, stored and written as Bbig row 16384+d = -2c.
// ---------------------------------------------------------------------------
__global__ void assemble_kernel(const float* __restrict__ Linv,
                                const float* __restrict__ mu_num,
                                const float* __restrict__ gsum,
                                float* __restrict__ Bbig, float* c_g) {
  int d = blockIdx.x, k = blockIdx.y, e = threadIdx.x;
  __shared__ float Ld[DD];
  const float* Lk = Linv + k * DD * DD;
  Ld[e] = (e >= d) ? Lk[d * DD + e] : 0.0f;
  __syncthreads();
  int lo = (d > e) ? d : e;
  const float* Le = Lk + e * DD;
  float v = 0.0f;
  for (int mm = lo; mm < DD; ++mm) v += Ld[mm] * Le[mm];
  Bbig[(d * DD + e) * KK + k] = v;
  float gs = gsum[k];
  float cpart = v * (mu_num[k * DD + e] / gs);
  float cd = block_reduce_sum(cpart);
  if (e == 0) {
    c_g[k * DD + d] = cd;
    Bbig[(DD * DD + d) * KK + k] = -2.0f * cd;
  }
}

// ---------------------------------------------------------------------------
// K4c: s_k = mu^T cov_inv mu, coef_k = phi_k * det(2*pi*Sigma)^(-1/4), pads.
// ---------------------------------------------------------------------------
__global__ void finalize_kernel(const float* __restrict__ c_g,
                                const float* __restrict__ mu_num,
                                const float* __restrict__ gsum,
                                const float* __restrict__ ldet_g,
                                float* __restrict__ Bbig, float* coef) {
  int k = blockIdx.x, d = threadIdx.x;
  float gs = gsum[k];
  float t = (mu_num[k * DD + d] / gs) * c_g[k * DD + d];
  float s = block_reduce_sum(t);
  if (d == 0) {
    Bbig[(DD * DD + DD + 0) * KK + k] = s;    // row 16512: constant term
    Bbig[(DD * DD + DD + 1) * KK + k] = 0.0f; // pad rows
    Bbig[(DD * DD + DD + 2) * KK + k] = 0.0f;
    Bbig[(DD * DD + DD + 3) * KK + k] = 0.0f;
    float phi = gs / (float)NN;
    // 1/sqrt(sigma_det), sigma_det = sqrt(det(2*pi*Sigma))
    coef[k] = phi * expf(-0.25f * ((float)DD * LN2PI + ldet_g[k]));
  }
}

// ---------------------------------------------------------------------------
// K5: q[n,k] = -0.5 (x'Bx - 2x'c + s) for 16 rows x all 16 k as ONE 16x16
// WMMA tile: contraction [x(x)x | x | 1] (len 16516) vs Bbig. 4 tiles/wave
// amortize the Bbig stream. Then per-row max/sum over k via shfl_xor.
// ---------------------------------------------------------------------------
#define NROWS_B (DD * DD + DD + 4)  // 16516, multiple of 4
__global__ void energy_kernel(const float* __restrict__ X,
                              const float* __restrict__ Bbig,
                              const float* __restrict__ coef,
                              const float* __restrict__ sig_diag,
                              float* __restrict__ out) {
  __shared__ float Xs[4][16][132];
  int lane = threadIdx.x;
  int m = lane & 15;
  int g2 = (lane >> 4) << 1;
#pragma unroll
  for (int t = 0; t < 4; ++t) {
    int n0 = (blockIdx.x * 4 + t) * 16;
    for (int idx = lane; idx < 16 * DD; idx += 32)
      Xs[t][idx >> 7][idx & 127] = X[(n0 + (idx >> 7)) * DD + (idx & 127)];
  }
  v8f acc0 = vzero8(), acc1 = vzero8(), acc2 = vzero8(), acc3 = vzero8();
  const float* x0 = &Xs[0][m][0];
  const float* x1 = &Xs[1][m][0];
  const float* x2 = &Xs[2][m][0];
  const float* x3 = &Xs[3][m][0];
  for (int j = 0; j < NROWS_B / 4; ++j) {
    int i0 = j * 4 + g2;
    int i1 = i0 + 1;
    v2f b;
    b.x = Bbig[i0 * KK + m];
    b.y = Bbig[i1 * KK + m];
    v2f a;
    // tile 0
    a.x = (i0 < DD * DD) ? x0[i0 >> 7] * x0[i0 & 127]
          : (i0 < DD * DD + DD) ? x0[i0 - DD * DD]
          : (i0 == DD * DD + DD) ? 1.0f : 0.0f;
    a.y = (i1 < DD * DD) ? x0[i1 >> 7] * x0[i1 & 127]
          : (i1 < DD * DD + DD) ? x0[i1 - DD * DD]
          : (i1 == DD * DD + DD) ? 1.0f : 0.0f;
    acc0 = wmma4(a, b, acc0);
    // tile 1
    a.x = (i0 < DD * DD) ? x1[i0 >> 7] * x1[i0 & 127]
          : (i0 < DD * DD + DD) ? x1[i0 - DD * DD]
          : (i0 == DD * DD + DD) ? 1.0f : 0.0f;
    a.y = (i1 < DD * DD) ? x1[i1 >> 7] * x1[i1 & 127]
          : (i1 < DD * DD + DD) ? x1[i1 - DD * DD]
          : (i1 == DD * DD + DD) ? 1.0f : 0.0f;
    acc1 = wmma4(a, b, acc1);
    // tile 2
    a.x = (i0 < DD * DD) ? x2[i0 >> 7] * x2[i0 & 127]
          : (i0 < DD * DD + DD) ? x2[i0 - DD * DD]
          : (i0 == DD * DD + DD) ? 1.0f : 0.0f;
    a.y = (i1 < DD * DD) ? x2[i1 >> 7] * x2[i1 & 127]
          : (i1 < DD * DD + DD) ? x2[i1 - DD * DD]
          : (i1 == DD * DD + DD) ? 1.0f : 0.0f;
    acc2 = wmma4(a, b, acc2);
    // tile 3
    a.x = (i0 < DD * DD) ? x3[i0 >> 7] * x3[i0 & 127]
          : (i0 < DD * DD + DD) ? x3[i0 - DD * DD]
          : (i0 == DD * DD + DD) ? 1.0f : 0.0f;
    a.y = (i1 < DD * DD) ? x3[i1 >> 7] * x3[i1 & 127]
          : (i1 < DD * DD + DD) ? x3[i1 - DD * DD]
          : (i1 == DD * DD + DD) ? 1.0f : 0.0f;
    acc3 = wmma4(a, b, acc3);
  }
  float cf = coef[m];
  float sd = *sig_diag;
  v8f accs[4] = {acc0, acc1, acc2, acc3};
#pragma unroll
  for (int t = 0; t < 4; ++t) {
    int n0 = (blockIdx.x * 4 + t) * 16;
#pragma unroll
    for (int r = 0; r < 8; ++r) {
      float q = -0.5f * accs[t][r];
      float mv = fmaxf(q, 0.0f);
      mv = fmaxf(mv, __shfl_xor(mv, 1, 32));
      mv = fmaxf(mv, __shfl_xor(mv, 2, 32));
      mv = fmaxf(mv, __shfl_xor(mv, 4, 32));
      mv = fmaxf(mv, __shfl_xor(mv, 8, 32));
      float ex = cf * expf(q - mv);
      ex += __shfl_xor(ex, 1, 32);
      ex += __shfl_xor(ex, 2, 32);
      ex += __shfl_xor(ex, 4, 32);
      ex += __shfl_xor(ex, 8, 32);
      if (m == 0)
        out[n0 + ((lane >> 4) << 3) + r] =
            0.2f * (-mv - logf(ex + EPSF)) + 0.02f * sd;
    }
  }
}

// ---------------------------------------------------------------------------
// Host launcher.  Workspace layout (floats):
//   gamma[N*K] | M2[K*D*D] | mu_num[K*D] | gsum[K] | Linv[K*D*D] |
//   c_g[K*D] | ldet[K] | Bbig[16516*K] | coef[K] | sigma_diag[1]
//   total ~= 7.4 MB
// ---------------------------------------------------------------------------
extern "C" void kernel_launch(void* const* d_in, const int* in_sizes, int n_in,
                              void* d_out, int out_size, void* d_ws,
                              size_t ws_size, hipStream_t stream) {
  const float* X = (const float*)d_in[0];
  const float* W1 = (const float*)d_in[1];
  const float* b1 = (const float*)d_in[2];
  const float* W2 = (const float*)d_in[3];
  const float* b2 = (const float*)d_in[4];
  float* out = (float*)d_out;

  float* ws = (float*)d_ws;
  float* gamma = ws;                 ws += (size_t)NN * KK;
  float* M2 = ws;                    ws += (size_t)KK * DD * DD;
  float* mu_num = ws;                ws += (size_t)KK * DD;
  float* gsum = ws;                  ws += KK;
  float* Linv = ws;                  ws += (size_t)KK * DD * DD;
  float* c_g = ws;                   ws += (size_t)KK * DD;
  float* ldet = ws;                  ws += KK;
  float* Bbig = ws;                  ws += (size_t)NROWS_B * KK;
  float* coef = ws;                  ws += KK;
  float* sig_diag = ws;              ws += 1;

  zero_kernel<<<(KK * DD * DD + 255) / 256, 256, 0, stream>>>(M2, sig_diag);
  gamma_kernel<<<NN / 16 / 4, 128, 0, stream>>>(X, W1, b1, W2, b2, gamma);
  gsum_kernel<<<KK, 256, 0, stream>>>(gamma, gsum);
  mu_kernel<<<KK * DD, 256, 0, stream>>>(gamma, X, mu_num);
  m2_kernel<<<dim3(DD / 16, DD / 16, KK * NSPLIT), 32, 0, stream>>>(gamma, X, M2);
  chol_kernel<<<KK, DD, 0, stream>>>(M2, mu_num, gsum, Linv, ldet, sig_diag);
  assemble_kernel<<<dim3(DD, KK), DD, 0, stream>>>(Linv, mu_num, gsum, Bbig, c_g);
  finalize_kernel<<<KK, DD, 0, stream>>>(c_g, mu_num, gsum, ldet, Bbig, coef);
  energy_kernel<<<NN / 16 / 4, 32, 0, stream>>>(X, Bbig, coef, sig_diag, out);
}